// RPNPostProcessor_13314398618287
// MI455X (gfx1250) — compile-verified
//
#include <hip/hip_runtime.h>
#include <math.h>

// ---------------- problem constants (match reference) ----------------
#define NIMG 8
#define AA 3
#define HH 272
#define WW 400
#define HW (HH * WW)          // 108800
#define KTOT (AA * HW)        // 326400
#define PRE_NMS 6000
#define POST_NMS 1000
#define NMS_THRESH 0.7f
#define BBOX_CLIP 4.135166556742356f   // log(1000/16)
#define NBINS 1024
#define SORTN 8192
#define TPB 1024
#define MASKW ((PRE_NMS + 31) / 32)    // 188

__device__ __forceinline__ unsigned orderedF32(float f) {
  unsigned u = __float_as_uint(f);
  // monotonic-increasing mapping float -> u32
  return (u & 0x80000000u) ? ~u : (u | 0x80000000u);
}

// ============================================================================
// Kernel 1: per-image fused sigmoid + exact top-6000 (sorted desc, idx-asc ties)
// One block per image.  LDS: 8192 u64 sort keys (64KB) + 1024-bin histogram.
// ============================================================================
extern "C" __global__ __launch_bounds__(TPB)
void rpn_topk_kernel(const float* __restrict__ obj,
                     int* __restrict__ topk_idx,
                     float* __restrict__ topk_score) {
  extern __shared__ unsigned char smem[];
  unsigned long long* keys = (unsigned long long*)smem;        // SORTN * 8
  int* hist = (int*)(smem + SORTN * 8);                        // NBINS * 4
  int* ctrl = (int*)(smem + SORTN * 8 + NBINS * 4);            // [0]=thresh bin, [1]=count

  const int n = blockIdx.x;
  const int tid = threadIdx.x;
  const float* base = obj + (size_t)n * KTOT;

  for (int b = tid; b < NBINS; b += TPB) hist[b] = 0;
  if (tid == 0) ctrl[1] = 0;
  __syncthreads();

  // pass 1: histogram of logits (coalesced over memory (a,h,w) order)
  for (int m = tid; m < KTOT; m += TPB) {
    float x = base[m];
    int b = (int)((x + 8.0f) * 64.0f);
    b = min(max(b, 0), NBINS - 1);
    atomicAdd(&hist[b], 1);
  }
  __syncthreads();

  // threshold bin: largest t with suffix-count >= PRE_NMS
  if (tid == 0) {
    int cum = 0, t = 0;
    for (int b = NBINS - 1; b >= 0; --b) {
      cum += hist[b];
      if (cum >= PRE_NMS) { t = b; break; }
    }
    ctrl[0] = t;
  }
  __syncthreads();
  const int t = ctrl[0];

  // pass 2: compact candidates into LDS as sortable 64-bit keys.
  // flat index k is in (h,w,a) order:  k = (h*W + w)*A + a   (permute(0,2,3,1))
  for (int m = tid; m < KTOT; m += TPB) {
    float x = base[m];
    int b = (int)((x + 8.0f) * 64.0f);
    b = min(max(b, 0), NBINS - 1);
    if (b >= t) {
      int pos = atomicAdd(&ctrl[1], 1);
      if (pos < SORTN) {
        int a = m / HW;
        int rest = m - a * HW;                 // h*W + w
        unsigned k = (unsigned)(rest * AA + a);
        keys[pos] = ((unsigned long long)orderedF32(x) << 32) |
                    (unsigned long long)(~k);  // ~k: idx-asc on score ties (desc sort)
      }
    }
  }
  __syncthreads();
  int num = ctrl[1];
  if (num > SORTN) num = SORTN;
  for (int i = num + tid; i < SORTN; i += TPB) keys[i] = 0ULL;  // pad sorts last
  __syncthreads();

  // in-LDS bitonic sort, descending
  for (unsigned kk = 2; kk <= SORTN; kk <<= 1) {
    for (unsigned j = kk >> 1; j > 0; j >>= 1) {
      for (unsigned i = tid; i < SORTN; i += TPB) {
        unsigned ix = i ^ j;
        if (ix > i) {
          unsigned long long a = keys[i], b = keys[ix];
          bool segUp = ((i & kk) == 0);
          bool doswap = segUp ? (a < b) : (a > b);
          if (doswap) { keys[i] = b; keys[ix] = a; }
        }
      }
      __syncthreads();
    }
  }

  // emit top PRE_NMS: recover idx and sigmoid(logit)
  for (int i = tid; i < PRE_NMS; i += TPB) {
    unsigned long long key = keys[i];
    unsigned idx = ~(unsigned)(key & 0xFFFFFFFFu);
    unsigned ord = (unsigned)(key >> 32);
    unsigned ub = (ord & 0x80000000u) ? (ord & 0x7FFFFFFFu) : ~ord;
    float logit = __uint_as_float(ub);
    topk_idx[n * PRE_NMS + i] = (int)idx;
    topk_score[n * PRE_NMS + i] = 1.0f / (1.0f + expf(-logit));
  }
}

// ============================================================================
// Kernel 2: gather deltas/anchors for the selected 48000 proposals, decode,
// clip to image, apply remove_small_boxes mask (score -> -inf).
// ============================================================================
extern "C" __global__ void rpn_decode_kernel(const float* __restrict__ breg,
                                             const float* __restrict__ anchors,
                                             const int* __restrict__ topk_idx,
                                             float* __restrict__ scores,
                                             float* __restrict__ boxes,
                                             const int* __restrict__ im_h,
                                             const int* __restrict__ im_w) {
  int gid = blockIdx.x * blockDim.x + threadIdx.x;
  if (gid >= NIMG * PRE_NMS) return;
  int n = gid / PRE_NMS;
  int k = topk_idx[gid];
  int a = k % AA;
  int hw = k / AA;                 // h*W + w
  int h = hw / WW;
  int w = hw - h * WW;
  // box_regression layout: [n][a*4+c][h][w]
  size_t dbase = (size_t)n * (AA * 4 * HW) + (size_t)(a * 4) * HW + (size_t)h * WW + w;
  float dx = breg[dbase];
  float dy = breg[dbase + HW];
  float dw = fminf(breg[dbase + 2 * (size_t)HW], BBOX_CLIP);
  float dh = fminf(breg[dbase + 3 * (size_t)HW], BBOX_CLIP);
  float4 anc = ((const float4*)anchors)[(size_t)n * KTOT + k];
  float aw = anc.z - anc.x + 1.0f;
  float ah = anc.w - anc.y + 1.0f;
  float cx = anc.x + 0.5f * aw;
  float cy = anc.y + 0.5f * ah;
  float pcx = dx * aw + cx, pcy = dy * ah + cy;
  float pw = expf(dw) * aw, ph = expf(dh) * ah;
  float x0 = pcx - 0.5f * pw, y0 = pcy - 0.5f * ph;
  float x1 = pcx + 0.5f * pw - 1.0f, y1 = pcy + 0.5f * ph - 1.0f;
  float mw = (float)im_w[0] - 1.0f, mh = (float)im_h[0] - 1.0f;
  x0 = fminf(fmaxf(x0, 0.0f), mw);
  x1 = fminf(fmaxf(x1, 0.0f), mw);
  y0 = fminf(fmaxf(y0, 0.0f), mh);
  y1 = fminf(fmaxf(y1, 0.0f), mh);
  ((float4*)boxes)[gid] = make_float4(x0, y0, x1, y1);
  bool keep = ((x1 - x0 + 1.0f) >= 0.0f) && ((y1 - y0 + 1.0f) >= 0.0f);
  if (!keep) scores[gid] = -INFINITY;
}

// ============================================================================
// Kernel 3: greedy NMS, one block per image.  Boxes AND scores staged into LDS
// with CDNA5 async global->LDS loads (ASYNCcnt / s_wait_asynccnt), so the 1000
// serial selection steps touch only LDS.  2 barriers/step via double-buffered
// argmin cell; monotone frontier (sorted scores => dead prefix) shrinks both
// the first-alive scan and the suppression range.
// LDS: 96000 (boxes) + 24000 (areas) + 24000 (scores) + 752 (mask) + ctrl.
// ============================================================================
extern "C" __global__ __launch_bounds__(TPB)
void rpn_nms_kernel(const float* __restrict__ boxes,
                    const float* __restrict__ scores,
                    float* __restrict__ out) {
  extern __shared__ unsigned char smem[];
  float4* sbox   = (float4*)smem;                                   //      0..96000
  float* sarea   = (float*)(smem + PRE_NMS * 16);                   //  96000..120000
  float* sscore  = (float*)(smem + PRE_NMS * 20);                   // 120000..144000
  unsigned* mask = (unsigned*)(smem + PRE_NMS * 24);                // 144000..144752
  int* s_best    = (int*)(smem + PRE_NMS * 24 + MASKW * 4);         // [2] double-buffer
  int* s_startw  = (int*)(smem + PRE_NMS * 24 + MASKW * 4 + 8);

  const int n = blockIdx.x;
  const int tid = threadIdx.x;
  const float4* gbox = (const float4*)boxes + (size_t)n * PRE_NMS;
  const float* gsc = scores + (size_t)n * PRE_NMS;

#if defined(__gfx1250__)
  // CDNA5 async copy engine: per-lane global -> LDS, tracked by ASYNCcnt.
  for (int j = tid; j < PRE_NMS; j += TPB) {
    unsigned ldsb = (unsigned)(unsigned long long)(void*)&sbox[j];
    const float4* gp = gbox + j;
    asm volatile("global_load_async_to_lds_b128 %0, %1, off"
                 :: "v"(ldsb), "v"(gp) : "memory");
    unsigned ldss = (unsigned)(unsigned long long)(void*)&sscore[j];
    const float* sp = gsc + j;
    asm volatile("global_load_async_to_lds_b32 %0, %1, off"
                 :: "v"(ldss), "v"(sp) : "memory");
  }
#else
  for (int j = tid; j < PRE_NMS; j += TPB) { sbox[j] = gbox[j]; sscore[j] = gsc[j]; }
#endif

  // zero this image's output slice while the async copies drain
  float* obox = out + (size_t)n * POST_NMS * 4;
  float* osc  = out + (size_t)NIMG * POST_NMS * 4 + (size_t)n * POST_NMS;
  float* oval = out + (size_t)NIMG * POST_NMS * 5 + (size_t)n * POST_NMS;
  for (int i = tid; i < POST_NMS * 4; i += TPB) obox[i] = 0.0f;
  for (int i = tid; i < POST_NMS; i += TPB) { osc[i] = 0.0f; oval[i] = 0.0f; }

  if (tid == 0) { s_best[0] = 0x7FFFFFFF; s_best[1] = 0x7FFFFFFF; *s_startw = 0; }

#if defined(__gfx1250__)
  asm volatile("s_wait_asynccnt 0" ::: "memory");
#endif
  __syncthreads();

  // areas + alive bitmask from LDS-resident data
  for (int j = tid; j < PRE_NMS; j += TPB) {
    float4 b = sbox[j];
    sarea[j] = (b.z - b.x + 1.0f) * (b.w - b.y + 1.0f);
  }
  for (int wrd = tid; wrd < MASKW; wrd += TPB) {
    unsigned m = 0;
    int jbase = wrd * 32;
    for (int b = 0; b < 32; ++b) {
      int j = jbase + b;
      if (j < PRE_NMS && sscore[j] > -INFINITY) m |= (1u << b);
    }
    mask[wrd] = m;
  }

  for (int sel = 0; sel < POST_NMS; ++sel) {
    __syncthreads();                       // A: prev suppression + slot reset visible
    const int sw = *s_startw;
    int* bestp = &s_best[sel & 1];
    for (int w = sw + tid; w < MASKW; w += TPB) {
      unsigned m = mask[w];
      if (m) atomicMin(bestp, w * 32 + (__ffs((int)m) - 1));
    }
    __syncthreads();                       // B: argmin complete
    const int i = *bestp;
    if (i == 0x7FFFFFFF) break;            // uniform; remainder stays zeroed
    if (tid == 0) {
      s_best[(sel + 1) & 1] = 0x7FFFFFFF;  // reset next slot (not in use this step)
      *s_startw = i >> 5;                  // prefix below i is permanently dead
      atomicAnd(&mask[i >> 5], ~(1u << (i & 31)));
      float4 bi0 = sbox[i];
      obox[sel * 4 + 0] = bi0.x;
      obox[sel * 4 + 1] = bi0.y;
      obox[sel * 4 + 2] = bi0.z;
      obox[sel * 4 + 3] = bi0.w;
      osc[sel] = sscore[i];
      oval[sel] = 1.0f;
    }
    // i is uniform: every thread reads the selected box straight from LDS
    const float4 bi = sbox[i];
    const float ai = sarea[i];
    for (int j = i + 1 + tid; j < PRE_NMS; j += TPB) {
      unsigned mword = mask[j >> 5];
      if (mword & (1u << (j & 31))) {
        float4 bj = sbox[j];
        float xx0 = fmaxf(bi.x, bj.x), yy0 = fmaxf(bi.y, bj.y);
        float xx1 = fminf(bi.z, bj.z), yy1 = fminf(bi.w, bj.w);
        float inter = fmaxf(xx1 - xx0 + 1.0f, 0.0f) * fmaxf(yy1 - yy0 + 1.0f, 0.0f);
        float iou = inter / (ai + sarea[j] - inter);
        if (iou > NMS_THRESH) atomicAnd(&mask[j >> 5], ~(1u << (j & 31)));
      }
    }
  }
}

// ============================================================================
extern "C" void kernel_launch(void* const* d_in, const int* in_sizes, int n_in,
                              void* d_out, int out_size, void* d_ws, size_t ws_size,
                              hipStream_t stream) {
  (void)in_sizes; (void)n_in; (void)out_size; (void)ws_size;
  const float* obj = (const float*)d_in[0];     // [N,A,H,W]
  const float* breg = (const float*)d_in[1];    // [N,A*4,H,W]
  const float* anchors = (const float*)d_in[2]; // [N,K,4]
  const int* im_h = (const int*)d_in[3];
  const int* im_w = (const int*)d_in[4];
  float* out = (float*)d_out;                   // boxes | scores | valid

  char* ws = (char*)d_ws;
  int* topk_idx     = (int*)ws;                                   // N*6000 i32
  float* topk_score = (float*)(ws + (size_t)NIMG * PRE_NMS * 4);  // N*6000 f32
  float* boxes      = (float*)(ws + (size_t)NIMG * PRE_NMS * 8);  // N*6000*4 f32 (16B aligned)

  size_t sh1 = (size_t)SORTN * 8 + (size_t)NBINS * 4 + 64;
  rpn_topk_kernel<<<NIMG, TPB, sh1, stream>>>(obj, topk_idx, topk_score);

  int tot = NIMG * PRE_NMS;
  rpn_decode_kernel<<<(tot + 255) / 256, 256, 0, stream>>>(
      breg, anchors, topk_idx, topk_score, boxes, im_h, im_w);

  size_t sh3 = (size_t)PRE_NMS * 24 + (size_t)MASKW * 4 + 64;
  rpn_nms_kernel<<<NIMG, TPB, sh3, stream>>>(boxes, topk_score, out);
}